// TaGSim_NodeUnlabel_68332929679866
// MI455X (gfx1250) — compile-verified
//
#include <hip/hip_runtime.h>
#include <math.h>

// ---------------------------------------------------------------------------
// TaGSim-style graph similarity for MI455X (gfx1250, wave32).
//
// Roofline: 2 x 256MB fp32 adjacency matrices, each streamed twice -> ~1GB of
// reads, ~44us floor at 23.3 TB/s HBM. The matvecs are pure streaming (one
// wave32 per row, float4 global_load_b128, shuffle reduce) — WMMA would only
// add layout traffic there. The tensor-network bilinear forms (e1^T W_k e2,
// 16 neurons, fp32) are real matrix ops and run on V_WMMA_F32_16X16X4_F32,
// one wave per neuron, on zero-padded operands so every load is unconditional.
// ---------------------------------------------------------------------------

typedef float v2f __attribute__((ext_vector_type(2)));
typedef float v8f __attribute__((ext_vector_type(8)));

// ---------------------------------------------------------------------------
// out[row] = sum_col A[row,col] * v[col]    (A is n x n, row-major)
// One wave32 per row; 8 rows per 256-thread block; 512B contiguous per wave
// per iteration -> saturates HBM.
// ---------------------------------------------------------------------------
__global__ void matvec_rowwave(const float* __restrict__ A,
                               const float* __restrict__ v,
                               float* __restrict__ out, int n) {
  const int lane = threadIdx.x & 31;
  const int wave = threadIdx.x >> 5;
  const int row  = blockIdx.x * (blockDim.x >> 5) + wave;
  if (row >= n) return;
  const float* __restrict__ arow = A + (size_t)row * (size_t)n;
  float acc = 0.f;
  const int nvec = n & ~127;  // multiple of 32 lanes * 4 floats
  for (int base = lane * 4; base < nvec; base += 128) {
    float4 a = *reinterpret_cast<const float4*>(arow + base);
    float4 x = *reinterpret_cast<const float4*>(v + base);
    acc += a.x * x.x + a.y * x.y + a.z * x.z + a.w * x.w;
  }
  for (int idx = nvec + lane; idx < n; idx += 32) acc += arow[idx] * v[idx];
  for (int off = 16; off > 0; off >>= 1) acc += __shfl_down(acc, off, 32);
  if (lane == 0) out[row] = acc;
}

// ---------------------------------------------------------------------------
// Clamped integer histogram (bin_onehot + sum-pool fused). Degree values are
// exact integers in fp32 (sums of 0/1 and of small ints, all < 2^24).
// ---------------------------------------------------------------------------
__global__ void hist_kernel(const float* __restrict__ vec, int n, int nbins,
                            float* __restrict__ out) {
  __shared__ float hcnt[64];
  if (threadIdx.x < 64) hcnt[threadIdx.x] = 0.f;
  __syncthreads();
  int i = blockIdx.x * blockDim.x + threadIdx.x;
  if (i < n) {
    int bin = (int)(vec[i] + 0.5f);
    bin = bin < 0 ? 0 : (bin >= nbins ? nbins - 1 : bin);
    atomicAdd(&hcnt[bin], 1.0f);
  }
  __syncthreads();
  if ((int)threadIdx.x < nbins) {
    float c = hcnt[threadIdx.x];
    if (c != 0.f) atomicAdd(&out[threadIdx.x], c);
  }
}

__global__ void zero_kernel(float* __restrict__ p, int n) {
  int i = blockIdx.x * blockDim.x + threadIdx.x;
  if (i < n) p[i] = 0.f;
}

// ---------------------------------------------------------------------------
// Repack W[D,D,K] (k innermost) into Wp[K][Dpad][Dpad], zero-padded, so the
// WMMA A-operand becomes a contiguous unconditional 8-byte load per lane.
// ---------------------------------------------------------------------------
__global__ void pad_weights_kernel(const float* __restrict__ W,
                                   float* __restrict__ Wp,
                                   int D, int Dpad, int K) {
  int idx = blockIdx.x * blockDim.x + threadIdx.x;
  int total = K * Dpad * Dpad;
  if (idx >= total) return;
  int k = idx / (Dpad * Dpad);
  int r = idx - k * (Dpad * Dpad);
  int d = r / Dpad;
  int e = r - d * Dpad;
  Wp[idx] = (d < D && e < D) ? W[((size_t)d * D + e) * K + k] : 0.f;
}

// ---------------------------------------------------------------------------
// Per-wave bilinear form s = e1^T Wp e2 using V_WMMA_F32_16X16X4_F32.
// Wp: this neuron's padded Dpad x Dpad tile (row-major). e1/e2: LDS vectors,
// zero-padded to Dpad. All loads unconditional; lane participation handled
// with a select mask (no exec divergence around the WMMAs).
//   A (16x4): lanes 0-15 hold cols {0,1}, lanes 16-31 cols {2,3}, row = lane%16
//   B (4x16): lanes 0-15 hold rows {0,1}, lanes 16-31 rows {2,3}, col = lane%16
//             only col n=0 is non-zero (carries e2 chunk)
//   C (16x16): col 0 sits in lane 0 (M=0..7, VGPR r=M) and lane 16 (M=8..15)
// ---------------------------------------------------------------------------
template <int Dpad>
__device__ __forceinline__ float wave_bilinear_wmma(const float* __restrict__ Wp,
                                                    const float* __restrict__ e1,
                                                    const float* __restrict__ e2) {
  const int lane = threadIdx.x & 31;
  const int m    = lane & 15;
  const int hi   = lane >> 4;       // 0: lanes 0-15, 1: lanes 16-31
  const int kk0  = hi * 2;          // A column pair / B row pair
  const float sel = (m == 0) ? 1.f : 0.f;  // lanes 0 and 16 carry column n=0
  float s = 0.f;
#pragma unroll
  for (int dt = 0; dt < Dpad; dt += 16) {
    v8f c = {};
    const float* __restrict__ arow = Wp + (size_t)(dt + m) * Dpad + kk0;
#pragma unroll
    for (int e0 = 0; e0 < Dpad; e0 += 4) {
      v2f a  = *reinterpret_cast<const v2f*>(arow + e0);            // global_load_b64
      v2f e2p = *reinterpret_cast<const v2f*>(&e2[e0 + kk0]);       // ds_load_b64
      v2f b;
      b.x = e2p.x * sel;
      b.y = e2p.y * sel;
      c = __builtin_amdgcn_wmma_f32_16x16x4_f32(
          /*neg_a=*/false, a, /*neg_b=*/false, b,
          /*c_mod=*/(short)0, c, /*reuse_a=*/false, /*reuse_b=*/false);
    }
    const int rbase = dt + hi * 8;
    float part = 0.f;
#pragma unroll
    for (int r = 0; r < 8; ++r) part += c[r] * e1[rbase + r];
    s += part * sel;  // only lanes 0 and 16 contribute real data
  }
  for (int off = 16; off > 0; off >>= 1) s += __shfl_down(s, off, 32);
  return s;  // valid on lane 0
}

__device__ __forceinline__ float relu_(float x) { return x > 0.f ? x : 0.f; }

// 16 -> 16 -> 8 -> 4 -> 1 head, relu/relu/relu/sigmoid. W stored [in,out].
__device__ float mlp4(const float* x0,
                      const float* __restrict__ W0, const float* __restrict__ B0,
                      const float* __restrict__ W1, const float* __restrict__ B1,
                      const float* __restrict__ W2, const float* __restrict__ B2,
                      const float* __restrict__ W3, const float* __restrict__ B3) {
  float h0[16], h1[8], h2[4];
  for (int o = 0; o < 16; ++o) {
    float s = B0[o];
    for (int i = 0; i < 16; ++i) s += x0[i] * W0[i * 16 + o];
    h0[o] = relu_(s);
  }
  for (int o = 0; o < 8; ++o) {
    float s = B1[o];
    for (int i = 0; i < 16; ++i) s += h0[i] * W1[i * 8 + o];
    h1[o] = relu_(s);
  }
  for (int o = 0; o < 4; ++o) {
    float s = B2[o];
    for (int i = 0; i < 8; ++i) s += h1[i] * W2[i * 4 + o];
    h2[o] = relu_(s);
  }
  float s = B3[0];
  for (int i = 0; i < 4; ++i) s += h2[i] * W3[i];
  return 1.f / (1.f + expf(-s));
}

// ---------------------------------------------------------------------------
// Final stage: assemble pooled feature vectors, run both tensor networks
// (WMMA, one wave per neuron k), both MLP heads, and score->ged->sim math.
// Single block of 512 threads = 16 wave32s.
// hist layout: [h10_g1(10) | h50_g1(50) | h10_g2(10) | h50_g2(50)]
// ---------------------------------------------------------------------------
__global__ void final_kernel(
    const float* __restrict__ emb1, const float* __restrict__ emb2, int n1, int n2,
    const float* __restrict__ avg_v,
    const float* __restrict__ wp_in, const float* __restrict__ wb_in, const float* __restrict__ b_in,
    const float* __restrict__ wp_ie, const float* __restrict__ wb_ie, const float* __restrict__ b_ie,
    const float* mi_W0, const float* mi_b0, const float* mi_W1, const float* mi_b1,
    const float* mi_W2, const float* mi_b2, const float* mi_W3, const float* mi_b3,
    const float* me_W0, const float* me_b0, const float* me_W1, const float* me_b1,
    const float* me_W2, const float* me_b2, const float* me_W3, const float* me_b3,
    const float* __restrict__ hist, float* __restrict__ out) {
  __shared__ float e_in1[16], e_in2[16], e_ie1[64], e_ie2[64];
  __shared__ float s_in[16], s_ie[16];
  __shared__ float r1[16], r2[16];

  const int tid  = threadIdx.x;
  const int lane = tid & 31;
  const int wave = tid >> 5;

  // --- sum-pool of emb columns (element 0 of the 11-dim vectors) ---
  float a1 = 0.f, a2 = 0.f;
  for (int i = tid; i < n1; i += blockDim.x) a1 += emb1[i];
  for (int i = tid; i < n2; i += blockDim.x) a2 += emb2[i];
  for (int off = 16; off > 0; off >>= 1) {
    a1 += __shfl_down(a1, off, 32);
    a2 += __shfl_down(a2, off, 32);
  }
  if (lane == 0) { r1[wave] = a1; r2[wave] = a2; }

  // --- zero-pad + fill feature vectors from histograms ---
  if (tid >= 11 && tid < 16) { e_in1[tid] = 0.f; e_in2[tid] = 0.f; }
  if (tid >= 60 && tid < 64) { e_ie1[tid] = 0.f; e_ie2[tid] = 0.f; }
  if (tid < 10) {
    e_in1[1 + tid] = hist[tid];       e_in2[1 + tid] = hist[60 + tid];
    e_ie1[tid]     = hist[tid];       e_ie2[tid]     = hist[60 + tid];
  }
  if (tid < 50) {
    e_ie1[10 + tid] = hist[10 + tid]; e_ie2[10 + tid] = hist[70 + tid];
  }
  __syncthreads();
  if (tid == 0) {
    float s1 = 0.f, s2 = 0.f;
    for (int w = 0; w < (int)(blockDim.x >> 5); ++w) { s1 += r1[w]; s2 += r2[w]; }
    e_in1[0] = s1;
    e_in2[0] = s2;
  }
  __syncthreads();

  // --- tensor networks: wave w computes neuron k = w for both nets ---
  {
    const int k = wave;  // 16 waves, 16 neurons
    float sraw_in = wave_bilinear_wmma<16>(wp_in + (size_t)k * 16 * 16, e_in1, e_in2);
    float sraw_ie = wave_bilinear_wmma<64>(wp_ie + (size_t)k * 64 * 64, e_ie1, e_ie2);
    if (lane == 0) {
      float blk = 0.f;
      for (int j = 0; j < 11; ++j) blk += wb_in[k * 22 + j]      * e_in1[j];
      for (int j = 0; j < 11; ++j) blk += wb_in[k * 22 + 11 + j] * e_in2[j];
      s_in[k] = relu_(sraw_in + blk + b_in[k]);
      blk = 0.f;
      for (int j = 0; j < 60; ++j) blk += wb_ie[k * 120 + j]      * e_ie1[j];
      for (int j = 0; j < 60; ++j) blk += wb_ie[k * 120 + 60 + j] * e_ie2[j];
      s_ie[k] = relu_(sraw_ie + blk + b_ie[k]);
    }
  }
  __syncthreads();

  // --- MLP heads + score math (tiny: one thread) ---
  if (tid == 0) {
    float score_in = mlp4(s_in, mi_W0, mi_b0, mi_W1, mi_b1, mi_W2, mi_b2, mi_W3, mi_b3);
    float score_ie = mlp4(s_ie, me_W0, me_b0, me_W1, me_b1, me_W2, me_b2, me_W3, me_b3);
    float avg = avg_v[0];
    float pre_ged = (-logf(score_in) - logf(score_ie)) * avg;
    float pre_sim = expf(-pre_ged / avg);
    out[0] = pre_sim;
    out[1] = pre_ged;
    out[2] = score_in;
    out[3] = score_ie;
  }
}

// ---------------------------------------------------------------------------
// kernel_launch
// Inputs (setup_inputs order, tuples flattened):
//  0 emb1  1 emb2  2 adj1  3 adj2  4 avg_v
//  5 w_in  6 wb_in 7 b_in  8 w_ie  9 wb_ie 10 b_ie
//  11..18 mlp_in (W0,b0,W1,b1,W2,b2,W3,b3)
//  19..26 mlp_ie (W0,b0,W1,b1,W2,b2,W3,b3)
// ---------------------------------------------------------------------------
extern "C" void kernel_launch(void* const* d_in, const int* in_sizes, int n_in,
                              void* d_out, int out_size, void* d_ws, size_t ws_size,
                              hipStream_t stream) {
  const float* emb1 = (const float*)d_in[0];
  const float* emb2 = (const float*)d_in[1];
  const float* adj1 = (const float*)d_in[2];
  const float* adj2 = (const float*)d_in[3];
  const float* avg_v = (const float*)d_in[4];
  const float* w_in  = (const float*)d_in[5];
  const float* wb_in = (const float*)d_in[6];
  const float* b_in  = (const float*)d_in[7];
  const float* w_ie  = (const float*)d_in[8];
  const float* wb_ie = (const float*)d_in[9];
  const float* b_ie  = (const float*)d_in[10];

  const int n1 = in_sizes[0];
  const int n2 = in_sizes[1];
  const int K  = 16;

  float* ws    = (float*)d_ws;
  float* h1g1  = ws;
  float* h2g1  = ws + (size_t)n1;
  float* h1g2  = ws + 2 * (size_t)n1;
  float* h2g2  = ws + 2 * (size_t)n1 + n2;
  float* hist  = ws + 2 * (size_t)n1 + 2 * (size_t)n2;          // 120 floats
  float* wp_in = hist + 128;                                    // K*16*16 = 4096
  float* wp_ie = wp_in + (size_t)K * 16 * 16;                   // K*64*64 = 65536

  // zero histograms (harness poisons d_ws; we own re-init every call)
  zero_kernel<<<1, 128, 0, stream>>>(hist, 120);

  // repack + zero-pad tensor-network weights for clean WMMA operand loads
  pad_weights_kernel<<<(K * 16 * 16 + 255) / 256, 256, 0, stream>>>(w_in, wp_in, 11, 16, K);
  pad_weights_kernel<<<(K * 64 * 64 + 255) / 256, 256, 0, stream>>>(w_ie, wp_ie, 60, 64, K);

  // degree propagation: h1 = A@emb, h2 = A@h1   (8 rows / 256-thread block)
  const int rpb = 256 / 32;
  matvec_rowwave<<<(n1 + rpb - 1) / rpb, 256, 0, stream>>>(adj1, emb1, h1g1, n1);
  matvec_rowwave<<<(n1 + rpb - 1) / rpb, 256, 0, stream>>>(adj1, h1g1, h2g1, n1);
  matvec_rowwave<<<(n2 + rpb - 1) / rpb, 256, 0, stream>>>(adj2, emb2, h1g2, n2);
  matvec_rowwave<<<(n2 + rpb - 1) / rpb, 256, 0, stream>>>(adj2, h1g2, h2g2, n2);

  // fused bin_onehot + sum-pool
  hist_kernel<<<(n1 + 255) / 256, 256, 0, stream>>>(h1g1, n1, 10, hist + 0);
  hist_kernel<<<(n1 + 255) / 256, 256, 0, stream>>>(h2g1, n1, 50, hist + 10);
  hist_kernel<<<(n2 + 255) / 256, 256, 0, stream>>>(h1g2, n2, 10, hist + 60);
  hist_kernel<<<(n2 + 255) / 256, 256, 0, stream>>>(h2g2, n2, 50, hist + 70);

  // tensor networks (WMMA) + MLP heads + score math
  final_kernel<<<1, 512, 0, stream>>>(
      emb1, emb2, n1, n2, avg_v,
      wp_in, wb_in, b_in, wp_ie, wb_ie, b_ie,
      (const float*)d_in[11], (const float*)d_in[12], (const float*)d_in[13], (const float*)d_in[14],
      (const float*)d_in[15], (const float*)d_in[16], (const float*)d_in[17], (const float*)d_in[18],
      (const float*)d_in[19], (const float*)d_in[20], (const float*)d_in[21], (const float*)d_in[22],
      (const float*)d_in[23], (const float*)d_in[24], (const float*)d_in[25], (const float*)d_in[26],
      hist, (float*)d_out);
}